// FOG_55301998903392
// MI455X (gfx1250) — compile-verified
//
#include <hip/hip_runtime.h>
#include <stdint.h>

typedef float v2f __attribute__((ext_vector_type(2)));
typedef float v8f __attribute__((ext_vector_type(8)));

#define TPB 256
#define IPT 16
#define TILE 4096
#define RDX 16
#define NWAVES 8

// ---------------------------------------------------------------------------
// 256-element inclusive scan on the matrix pipe: V_WMMA_F32_16X16X4_F32.
// totals[256] (LDS, f32 integers < 2^24, exact) -> S[256] inclusive scan.
// X[r][c] = totals[r*16+c].  R = X*U (U[k][c]=k<=c), F = L*Trow + R (L[r][k]=k<r).
// All waves execute redundantly (uniform control flow, EXEC all ones).
// ---------------------------------------------------------------------------
__device__ __forceinline__ void wmma_scan256(int tid, const float* totals,
                                             float* Trow, float* S) {
  const int lane = tid & 31;
  const int ls = lane & 15;       // row (A) / col (B,D)
  const int hi = lane >> 4;       // K-half select
  v8f acc = {0.f, 0.f, 0.f, 0.f, 0.f, 0.f, 0.f, 0.f};

  // Step 1: row-wise inclusive scan  R = X * U  (K=16 as 4 chained K=4 WMMAs)
#pragma unroll
  for (int j = 0; j < 4; ++j) {
    const int k0 = 4 * j + 2 * hi;
    v2f a, bm;
    a[0] = totals[ls * 16 + k0];
    a[1] = totals[ls * 16 + k0 + 1];
    bm[0] = (k0 <= ls) ? 1.f : 0.f;
    bm[1] = (k0 + 1 <= ls) ? 1.f : 0.f;
    acc = __builtin_amdgcn_wmma_f32_16x16x4_f32(false, a, false, bm, (short)0,
                                                acc, false, false);
  }
  // Row totals T[r] = R[r][15] live in lane 15 (rows 0-7) / lane 31 (rows 8-15)
  if (lane == 15) {
#pragma unroll
    for (int i = 0; i < 8; ++i) Trow[i] = acc[i];
  }
  if (lane == 31) {
#pragma unroll
    for (int i = 0; i < 8; ++i) Trow[8 + i] = acc[i];
  }
  __syncthreads();
  // Step 2: add exclusive-scan of row totals:  F = L * Y + R, Y[k][c] = T[k]
#pragma unroll
  for (int j = 0; j < 4; ++j) {
    const int k0 = 4 * j + 2 * hi;
    v2f a, bm;
    a[0] = (k0 < ls) ? 1.f : 0.f;
    a[1] = (k0 + 1 < ls) ? 1.f : 0.f;
    bm[0] = Trow[k0];
    bm[1] = Trow[k0 + 1];
    acc = __builtin_amdgcn_wmma_f32_16x16x4_f32(false, a, false, bm, (short)0,
                                                acc, false, false);
  }
  // D layout: VGPR i, lanes 0-15 -> M=i; lanes 16-31 -> M=8+i
#pragma unroll
  for (int i = 0; i < 8; ++i) S[(hi * 8 + i) * 16 + ls] = acc[i];
  __syncthreads();
}

// ---------------------------------------------------------------------------
// Hierarchical scan building block: each block scans TILE=4096 f32 elements
// (16/thread serial + WMMA block scan of 256 thread-totals), writes inclusive
// result and optional per-block sums.  Full tiles use b128 vector memops.
// ---------------------------------------------------------------------------
__global__ void scan4096_kernel(const float* __restrict__ in,
                                float* __restrict__ out, int n,
                                float* __restrict__ bsums) {
  __shared__ float totals[TPB];
  __shared__ float Trow[16];
  __shared__ float S[TPB];
  const int tid = threadIdx.x;
  const int base = blockIdx.x * TILE + tid * IPT;
  float x[IPT];
  float run = 0.f;
  const bool full = (base + IPT <= n);
  if (full) {
    const float4* p = (const float4*)(in + base);
#pragma unroll
    for (int q = 0; q < 4; ++q) {
      const float4 v = p[q];
      run += v.x; x[q * 4 + 0] = run;
      run += v.y; x[q * 4 + 1] = run;
      run += v.z; x[q * 4 + 2] = run;
      run += v.w; x[q * 4 + 3] = run;
    }
  } else {
#pragma unroll
    for (int i = 0; i < IPT; ++i) {
      float v = (base + i < n) ? in[base + i] : 0.f;
      run += v;
      x[i] = run;
    }
  }
  totals[tid] = run;
  __syncthreads();
  wmma_scan256(tid, totals, Trow, S);
  const float excl = S[tid] - totals[tid];
  if (full) {
    float4* p = (float4*)(out + base);
#pragma unroll
    for (int q = 0; q < 4; ++q) {
      float4 v;
      v.x = x[q * 4 + 0] + excl;
      v.y = x[q * 4 + 1] + excl;
      v.z = x[q * 4 + 2] + excl;
      v.w = x[q * 4 + 3] + excl;
      p[q] = v;
    }
  } else {
#pragma unroll
    for (int i = 0; i < IPT; ++i) {
      if (base + i < n) out[base + i] = x[i] + excl;
    }
  }
  if (bsums != nullptr && tid == TPB - 1) bsums[blockIdx.x] = S[TPB - 1];
}

__global__ void scan_addback_kernel(float* __restrict__ data,
                                    const float* __restrict__ bsumsIncl,
                                    int n) {
  const int b = blockIdx.x;
  if (b == 0) return;
  const float add = bsumsIncl[b - 1];
  const int base = b * TILE + threadIdx.x * IPT;
  if (base + IPT <= n) {
    float4* p = (float4*)(data + base);
#pragma unroll
    for (int q = 0; q < 4; ++q) {
      float4 v = p[q];
      v.x += add; v.y += add; v.z += add; v.w += add;
      p[q] = v;
    }
  } else {
#pragma unroll
    for (int i = 0; i < IPT; ++i) {
      if (base + i < n) data[base + i] += add;
    }
  }
}

// ---------------------------------------------------------------------------
// Problem-specific kernels
// ---------------------------------------------------------------------------
__global__ void init_out_kernel(float* __restrict__ out, int n4, int ntot) {
  int i = blockIdx.x * blockDim.x + threadIdx.x;
  if (i < ntot) out[i] = (i < n4) ? -1.f : 0.f;
}

__global__ void make_keys_kernel(const int* __restrict__ coords,
                                 const float* __restrict__ kern,
                                 uint32_t* __restrict__ keys,
                                 float* __restrict__ vals, int n) {
  int i = blockIdx.x * blockDim.x + threadIdx.x;
  if (i >= n) return;
  const int4 c = ((const int4*)coords)[i];  // one global_load_b128 per point
  const int off = (c.y & 1) + 2 * (c.z & 1) + 4 * (c.w & 1);
  vals[i] = (float)(1 << off) * kern[off];
  keys[i] = ((uint32_t)c.x << 27) | ((uint32_t)(c.w >> 1) << 18) |
            ((uint32_t)(c.z >> 1) << 9) | (uint32_t)(c.y >> 1);
}

__global__ void radix_count_kernel(const uint32_t* __restrict__ keys,
                                   float* __restrict__ histA, int n, int shift,
                                   int nb) {
  __shared__ uint32_t cnt[RDX];
  const int tid = threadIdx.x;
  const int b = blockIdx.x;
  if (tid < RDX) cnt[tid] = 0u;
  __syncthreads();
  const int base = b * TILE;
  for (int c = 0; c < IPT; ++c) {
    int i = base + c * TPB + tid;
    if (c + 1 < IPT && i + TPB < n) __builtin_prefetch(&keys[i + TPB], 0, 1);
    if (i < n) {
      uint32_t d = (keys[i] >> shift) & (RDX - 1u);
      atomicAdd(&cnt[d], 1u);
    }
  }
  __syncthreads();
  if (tid < RDX) histA[tid * nb + b] = (float)cnt[tid];  // digit-major
}

// Stable per-pass scatter: wave32 ballot match -> in-wave rank; per-wave digit
// counts -> cross-wave prefix; running totals order the 16 chunks.
__global__ void radix_scatter_kernel(const uint32_t* __restrict__ keys_in,
                                     const float* __restrict__ vals_in,
                                     uint32_t* __restrict__ keys_out,
                                     float* __restrict__ vals_out,
                                     const float* __restrict__ histInc,
                                     const float* __restrict__ histRaw, int n,
                                     int shift, int nb) {
  __shared__ uint32_t running[RDX];
  __shared__ uint32_t totals[RDX];
  __shared__ uint32_t waveCnt[NWAVES * RDX];
  __shared__ uint32_t wavePre[NWAVES * RDX];
  __shared__ uint32_t baseS[RDX];
  const int tid = threadIdx.x;
  const int b = blockIdx.x;
  const int lane = tid & 31;
  const int wv = tid >> 5;
  if (tid < RDX) {
    const int idx = tid * nb + b;
    baseS[tid] = (uint32_t)(histInc[idx] - histRaw[idx]);  // exclusive base
    running[tid] = 0u;
  }
  __syncthreads();
  const int tileBase = b * TILE;
  for (int c = 0; c < IPT; ++c) {
    const int i = tileBase + c * TPB + tid;
    if (c + 1 < IPT && i + TPB < n) {
      __builtin_prefetch(&keys_in[i + TPB], 0, 1);  // global_prefetch_b8
      __builtin_prefetch(&vals_in[i + TPB], 0, 1);
    }
    const bool ok = (i < n);
    uint32_t key = 0u;
    float v = 0.f;
    if (ok) {
      key = keys_in[i];
      v = vals_in[i];
    }
    const uint32_t d = (key >> shift) & (RDX - 1u);
    uint32_t m = __builtin_amdgcn_ballot_w32(ok);
#pragma unroll
    for (int bit = 0; bit < 4; ++bit) {
      uint32_t bb = __builtin_amdgcn_ballot_w32(((d >> bit) & 1u) != 0u);
      m &= ((d >> bit) & 1u) ? bb : ~bb;
    }
    if (tid < NWAVES * RDX) waveCnt[tid] = 0u;
    __syncthreads();
    const uint32_t ltmask = (lane == 0) ? 0u : (0xFFFFFFFFu >> (32 - lane));
    const uint32_t rankw = (uint32_t)__popc(m & ltmask);
    if (ok && rankw == 0u) waveCnt[wv * RDX + d] = (uint32_t)__popc(m);
    __syncthreads();
    if (tid < RDX) {
      uint32_t s = 0u;
#pragma unroll
      for (int w = 0; w < NWAVES; ++w) {
        wavePre[w * RDX + tid] = s;
        s += waveCnt[w * RDX + tid];
      }
      totals[tid] = s;
    }
    __syncthreads();
    if (ok) {
      const uint32_t rank = running[d] + wavePre[wv * RDX + d] + rankw;
      const uint32_t pos = baseS[d] + rank;
      keys_out[pos] = key;
      vals_out[pos] = v;
    }
    __syncthreads();
    if (tid < RDX) running[tid] += totals[tid];
    __syncthreads();
  }
}

__global__ void flags_kernel(const uint32_t* __restrict__ keys,
                             float* __restrict__ flagf, int n) {
  int i = blockIdx.x * blockDim.x + threadIdx.x;
  if (i >= n) return;
  flagf[i] = (i == 0 || keys[i] != keys[i - 1]) ? 1.f : 0.f;
}

__global__ void emit_kernel(const uint32_t* __restrict__ keys,
                            const float* __restrict__ vals,
                            const float* __restrict__ uidx,
                            float* __restrict__ out, int n) {
  int i = blockIdx.x * blockDim.x + threadIdx.x;
  if (i >= n) return;
  const uint32_t key = keys[i];
  const uint32_t u = (uint32_t)uidx[i] - 1u;  // inclusive-scan -> 0-based slot
  atomicAdd(&out[4u * (uint32_t)n + u], vals[i]);  // feats region (sums exact)
  const bool boundary = (i == 0) || (key != keys[i - 1]);
  if (boundary) {
    float4 row;
    row.x = (float)(key >> 27);           // b
    row.y = (float)(key & 511u);          // px
    row.z = (float)((key >> 9) & 511u);   // py
    row.w = (float)((key >> 18) & 511u);  // pz
    ((float4*)out)[u] = row;  // one global_store_b128 per unique voxel
  }
}

// ---------------------------------------------------------------------------
extern "C" void kernel_launch(void* const* d_in, const int* in_sizes, int n_in,
                              void* d_out, int out_size, void* d_ws,
                              size_t ws_size, hipStream_t stream) {
  const int* coords = (const int*)d_in[0];
  const float* kern = (const float*)d_in[1];
  float* out = (float*)d_out;
  const int N = in_sizes[0] / 4;
  const int NB = (N + TILE - 1) / TILE;       // sort tiles
  const int HSZ = RDX * NB;                   // histogram length
  const int NHB = (HSZ + TILE - 1) / TILE;    // hist-scan tiles

  char* w = (char*)d_ws;
  auto carve = [&](size_t bytes) {
    void* p = (void*)w;
    w += (bytes + 255) & ~(size_t)255;
    return p;
  };
  uint32_t* keys_a = (uint32_t*)carve((size_t)N * 4);
  float* vals_a = (float*)carve((size_t)N * 4);
  uint32_t* keys_b = (uint32_t*)carve((size_t)N * 4);
  float* vals_b = (float*)carve((size_t)N * 4);
  float* flagf = (float*)carve((size_t)N * 4);
  float* histA = (float*)carve((size_t)HSZ * 4);
  float* histB = (float*)carve((size_t)HSZ * 4);
  float* sums1 = (float*)carve((size_t)TILE * 4);
  (void)ws_size;
  (void)n_in;
  (void)out_size;

  const int nTot = 5 * N;
  init_out_kernel<<<(nTot + TPB - 1) / TPB, TPB, 0, stream>>>(out, 4 * N, nTot);
  make_keys_kernel<<<(N + TPB - 1) / TPB, TPB, 0, stream>>>(coords, kern,
                                                            keys_a, vals_a, N);

  uint32_t* kin = keys_a;
  float* vin = vals_a;
  uint32_t* kout = keys_b;
  float* vout = vals_b;
  for (int p = 0; p < 8; ++p) {  // 8 x 4-bit passes cover the 29-bit key
    const int shift = 4 * p;
    radix_count_kernel<<<NB, TPB, 0, stream>>>(kin, histA, N, shift, NB);
    scan4096_kernel<<<NHB, TPB, 0, stream>>>(histA, histB, HSZ, sums1);
    scan4096_kernel<<<1, TPB, 0, stream>>>(sums1, sums1, NHB, nullptr);
    scan_addback_kernel<<<NHB, TPB, 0, stream>>>(histB, sums1, HSZ);
    radix_scatter_kernel<<<NB, TPB, 0, stream>>>(kin, vin, kout, vout, histB,
                                                 histA, N, shift, NB);
    uint32_t* tk = kin; kin = kout; kout = tk;
    float* tv = vin; vin = vout; vout = tv;
  }
  // 8 passes -> sorted data back in keys_a / vals_a (== kin/vin)

  flags_kernel<<<(N + TPB - 1) / TPB, TPB, 0, stream>>>(kin, flagf, N);
  scan4096_kernel<<<NB, TPB, 0, stream>>>(flagf, flagf, N, sums1);
  scan4096_kernel<<<1, TPB, 0, stream>>>(sums1, sums1, NB, nullptr);
  scan_addback_kernel<<<NB, TPB, 0, stream>>>(flagf, sums1, N);

  emit_kernel<<<(N + TPB - 1) / TPB, TPB, 0, stream>>>(kin, vin, flagf, out, N);
}